// MultiPathConvolution_3667902071300
// MI455X (gfx1250) — compile-verified
//
#include <hip/hip_runtime.h>
#include <hip/hip_bf16.h>

// CDNA5 / gfx1250, wave32.
typedef float v2f __attribute__((ext_vector_type(2)));
typedef float v8f __attribute__((ext_vector_type(8)));

#define IN_DIM   128
#define ED_DIM   16
#define F_DIM    144            // 128 + 16
#define P_DIM    3
#define OUT_DIM  128
#define TILE_E   64             // 4 M-tiles of 16 edges
#define M_TILES  4
#define KP_STEPS 18             // 36 K=4 steps, processed in pairs
#define LDS_STRIDE 148          // 144 + 4 pad -> conflict-free column reads
#define WT_ELEMS (P_DIM * KP_STEPS * 2 * OUT_DIM)   // float4 count = 13824
#define WT_BYTES (WT_ELEMS * 16)                    // 221184 bytes

__device__ __forceinline__ void async_copy_b128(unsigned lds_off, unsigned long long gaddr) {
    // GLOBAL_LOAD_ASYNC_TO_LDS_B128: per-lane 16B memory -> LDS, tracked by ASYNCcnt.
    asm volatile("global_load_async_to_lds_b128 %0, %1, off"
                 :: "v"(lds_off), "v"(gaddr) : "memory");
}

template <bool USE_WT>
__global__ __launch_bounds__(256) void mpconv_wmma_kernel(
    const float* __restrict__ x,           // [N, 128]
    const int*   __restrict__ edge_index,  // [2, E]
    const float* __restrict__ edge_attr,   // [E, 16]
    const float* __restrict__ edge_time,   // [E]
    const float* __restrict__ current_time,// [1]
    const float* __restrict__ W,           // [3, 144, 128]
    const float4* __restrict__ Wt,         // fragment-major repack (or null)
    const float* __restrict__ b,           // [3, 128]
    const float* __restrict__ decay,       // [3]
    float*       __restrict__ out,         // [N, 3, 128]
    int E)
{
    __shared__ float s_comb[TILE_E * LDS_STRIDE];
    __shared__ float s_tw[TILE_E][P_DIM];
    __shared__ int   s_dst[TILE_E];

    const int tid  = threadIdx.x;
    const int e0   = blockIdx.x * TILE_E;
    const int lane = tid & 31;
    const int hi   = lane >> 4;   // lane half: selects K pair within 16x4 A layout
    const int lm   = lane & 15;
    // Force the wave id into an SGPR: makes edge_index row reads scalar (s_load,
    // KMcnt, clause-batched) and gives Wt/b loads a scalar base address.
    const int wave = __builtin_amdgcn_readfirstlane(tid >> 5);

    // ---- Async stage: x[src] rows (wave w owns rows 8w..8w+7; 512B per row) ----
    #pragma unroll
    for (int i = 0; i < 8; ++i) {
        const int r = wave * 8 + i;
        const int e = e0 + r;                            // wave-uniform
        int src = 0;
        if (e < E) src = edge_index[e];                  // scalar load, scalar branch
        unsigned long long ga =
            (unsigned long long)(const void*)(x + (size_t)src * IN_DIM) + (unsigned)(lane * 16);
        unsigned lo =
            (unsigned)(unsigned long long)(const void*)&s_comb[r * LDS_STRIDE] + (unsigned)(lane * 16);
        async_copy_b128(lo, ga);
    }
    // ---- Async stage: edge_attr block (4096B contiguous; wave w owns 512B) ----
    {
        const int fi = wave * 128 + lane * 4;            // float index in tile
        const int r  = fi >> 4;
        const int c  = fi & 15;
        size_t gidx = (size_t)e0 * ED_DIM + (size_t)fi;
        if ((e0 + r) >= E) gidx = 0;                     // clamp OOB tail
        unsigned long long ga = (unsigned long long)(const void*)(edge_attr + gidx);
        unsigned lo =
            (unsigned)(unsigned long long)(const void*)&s_comb[r * LDS_STRIDE + 128 + c];
        async_copy_b128(lo, ga);
    }
    asm volatile("s_wait_asynccnt 0" ::: "memory");

    // ---- Temporal weights + destination ids (pow2 mapping: row=tid>>2, p=tid&3) ----
    {
        const int row = tid >> 2;
        const int pq  = tid & 3;
        const int e   = e0 + row;
        if (pq < P_DIM) {
            float tw = 0.f;
            if (e < E) tw = __expf(-decay[pq] * (current_time[0] - edge_time[e]));
            s_tw[row][pq] = tw;
        } else {
            s_dst[row] = (e < E) ? edge_index[E + e] : 0;
        }
    }
    __syncthreads();

    // Each wave owns 3 of the 24 sixteen-wide column tiles of the 384-wide output.
    v8f acc[M_TILES][3];
    int pth[3], cbase[3];
    #pragma unroll
    for (int j = 0; j < 3; ++j) {
        const int nt = wave * 3 + j;                     // scalar
        pth[j]   = nt >> 3;                              // 8 column tiles per path
        cbase[j] = (nt & 7) * 16;
        const float bv = b[pth[j] * OUT_DIM + cbase[j] + lm];  // bias folded into init
        v8f c;
        #pragma unroll
        for (int v = 0; v < 8; ++v) c[v] = bv;
        #pragma unroll
        for (int mt = 0; mt < M_TILES; ++mt) acc[mt][j] = c;
    }

    // ---- Main loop: 18 K-pairs; per pair: 3 B-frag b128 loads, 8 A ds_b64, 24 WMMA ----
    for (int kp = 0; kp < KP_STEPS; ++kp) {
        float4 bf[3];
        #pragma unroll
        for (int j = 0; j < 3; ++j) {
            if (USE_WT) {
                bf[j] = Wt[((pth[j] * KP_STEPS + kp) * 2 + hi) * OUT_DIM + cbase[j] + lm];
            } else {
                const float* wp =
                    W + ((size_t)pth[j] * F_DIM + 8 * kp + 2 * hi) * OUT_DIM + cbase[j] + lm;
                bf[j].x = wp[0];
                bf[j].y = wp[OUT_DIM];
                bf[j].z = wp[4 * OUT_DIM];
                bf[j].w = wp[5 * OUT_DIM];
            }
        }
        #pragma unroll
        for (int mt = 0; mt < M_TILES; ++mt) {
            const float* ap = &s_comb[(mt * 16 + lm) * LDS_STRIDE + 8 * kp + 2 * hi];
            v2f a0; a0.x = ap[0]; a0.y = ap[1];   // K-step 2*kp
            v2f a1; a1.x = ap[4]; a1.y = ap[5];   // K-step 2*kp+1
            #pragma unroll
            for (int j = 0; j < 3; ++j) {
                v2f b0; b0.x = bf[j].x; b0.y = bf[j].y;
                acc[mt][j] = __builtin_amdgcn_wmma_f32_16x16x4_f32(
                    false, a0, false, b0, (short)0, acc[mt][j], false, false);
            }
            #pragma unroll
            for (int j = 0; j < 3; ++j) {
                v2f b1; b1.x = bf[j].z; b1.y = bf[j].w;
                acc[mt][j] = __builtin_amdgcn_wmma_f32_16x16x4_f32(
                    false, a1, false, b1, (short)0, acc[mt][j], false, false);
            }
        }
    }

    // ---- Epilogue: temporal scaling + scatter atomic-add ----
    #pragma unroll
    for (int mt = 0; mt < M_TILES; ++mt) {
        #pragma unroll
        for (int j = 0; j < 3; ++j) {
            const int p  = pth[j];
            const int cb = cbase[j];
            #pragma unroll
            for (int v = 0; v < 8; ++v) {
                const int m = mt * 16 + v + hi * 8;  // C/D layout: VGPR v -> row v / v+8
                const int e = e0 + m;
                if (e < E) {
                    const float val = acc[mt][j][v] * s_tw[m][p];
                    const int d = s_dst[m];
                    atomicAdd(out + ((size_t)d * P_DIM + p) * OUT_DIM + cb + lm, val);
                }
            }
        }
    }
}

// Repack W [3,144,128] -> Wt fragment-major: Wt[((p*18+kp)*2+hi)*128 + col] =
// { W[p][8kp+2hi][col], W[p][8kp+2hi+1][col], W[p][8kp+4+2hi][col], W[p][8kp+4+2hi+1][col] }
__global__ __launch_bounds__(256) void transform_w_kernel(const float* __restrict__ W,
                                                          float4* __restrict__ Wt) {
    const int i = blockIdx.x * 256 + threadIdx.x;
    if (i >= WT_ELEMS) return;
    const int col = i & 127;
    int t = i >> 7;
    const int hi = t & 1; t >>= 1;
    const int kp = t % KP_STEPS;
    const int p  = t / KP_STEPS;
    const float* wp = W + ((size_t)p * F_DIM + 8 * kp + 2 * hi) * OUT_DIM + col;
    float4 v;
    v.x = wp[0];
    v.y = wp[OUT_DIM];
    v.z = wp[4 * OUT_DIM];
    v.w = wp[5 * OUT_DIM];
    Wt[i] = v;
}

__global__ void zero_out_kernel(float* __restrict__ p, long long n) {
    long long i = (long long)blockIdx.x * blockDim.x + threadIdx.x;
    if (i < n) p[i] = 0.f;
}

extern "C" void kernel_launch(void* const* d_in, const int* in_sizes, int n_in,
                              void* d_out, int out_size, void* d_ws, size_t ws_size,
                              hipStream_t stream) {
    const float* x            = (const float*)d_in[0];
    const int*   edge_index   = (const int*)  d_in[1];
    const float* edge_attr    = (const float*)d_in[2];
    const float* edge_time    = (const float*)d_in[3];
    const float* current_time = (const float*)d_in[4];
    const float* W            = (const float*)d_in[5];
    const float* b            = (const float*)d_in[6];
    const float* decay        = (const float*)d_in[7];
    float* out = (float*)d_out;

    const int E = in_sizes[3];            // edge_time is [E]

    // Zero the scatter-add target (harness poisons d_out).
    {
        long long n = (long long)out_size;
        int blocks = (int)((n + 255) / 256);
        zero_out_kernel<<<blocks, 256, 0, stream>>>(out, n);
    }

    const int tiles = (E + TILE_E - 1) / TILE_E;

    if (ws_size >= (size_t)WT_BYTES) {
        float4* Wt = (float4*)d_ws;
        transform_w_kernel<<<(WT_ELEMS + 255) / 256, 256, 0, stream>>>(W, Wt);
        mpconv_wmma_kernel<true><<<tiles, 256, 0, stream>>>(
            x, edge_index, edge_attr, edge_time, current_time, W, Wt, b, decay, out, E);
    } else {
        mpconv_wmma_kernel<false><<<tiles, 256, 0, stream>>>(
            x, edge_index, edge_attr, edge_time, current_time, W, nullptr, b, decay, out, E);
    }
}